// Net_85186381349659
// MI455X (gfx1250) — compile-verified
//
#include <hip/hip_runtime.h>
#include <math.h>

#define NB    8
#define NPER  8192
#define NTOT  (NB * NPER)      // 65536
#define ETOT  (16 * NTOT)      // 1048576
#define CW    64

typedef __attribute__((ext_vector_type(2))) float v2f;
typedef __attribute__((ext_vector_type(8))) float v8f;

// ---------------- utility kernels ----------------
__global__ void fill_f32(float* p, float v, int n) {
    int i = blockIdx.x * blockDim.x + threadIdx.x;
    if (i < n) p[i] = v;
}
__global__ void fill_i32(int* p, int v, int n) {
    int i = blockIdx.x * blockDim.x + threadIdx.x;
    if (i < n) p[i] = v;
}

__device__ __forceinline__ unsigned f2ord(float f) {
    unsigned u = __float_as_uint(f);
    return (u & 0x80000000u) ? ~u : (u | 0x80000000u);
}

// ---------------- SAGE layer 1 (input dim 2) ----------------
__global__ void edge_scatter_dim2(const int* __restrict__ src, const int* __restrict__ dst,
                                  const float* __restrict__ x, float* s, float* cnt, int ne) {
    int e = blockIdx.x * blockDim.x + threadIdx.x;
    if (e >= ne) return;
    int a = src[e], d = dst[e];
    atomicAdd(&s[d * 2 + 0], x[a * 2 + 0]);
    atomicAdd(&s[d * 2 + 1], x[a * 2 + 1]);
    atomicAdd(&cnt[d], 1.f);
}

__global__ void sage1_kernel(const float* __restrict__ x, const float* __restrict__ s,
                             const float* __restrict__ cnt,
                             const float* __restrict__ w1l, const float* __restrict__ w1r,
                             const float* __restrict__ b1, float* __restrict__ h, int n) {
    int t = blockIdx.x * blockDim.x + threadIdx.x;
    if (t >= n * CW) return;
    int i = t >> 6, c = t & 63;
    float cc = fmaxf(cnt[i], 1.f);
    float m0 = s[i * 2 + 0] / cc, m1 = s[i * 2 + 1] / cc;
    float v = m0 * w1l[c] + m1 * w1l[64 + c] + x[i * 2 + 0] * w1r[c] + x[i * 2 + 1] * w1r[64 + c] + b1[c];
    h[t] = fmaxf(v, 0.f);
}

// ---------------- TopK pooling machinery ----------------
__global__ void vec_norm64(const float* __restrict__ p, float* out) {
    __shared__ float sm[64];
    int t = threadIdx.x;
    sm[t] = p[t] * p[t];
    __syncthreads();
    if (t == 0) {
        float s = 0.f;
        for (int i = 0; i < 64; ++i) s += sm[i];
        out[0] = sqrtf(s);
    }
}

__global__ void score_kernel(const float* __restrict__ h, const float* __restrict__ p,
                             const float* __restrict__ nrm, float* score, unsigned* key, int n) {
    int i = blockIdx.x * blockDim.x + threadIdx.x;
    if (i >= n) return;
    const float* hp = h + (size_t)i * CW;
    float acc = 0.f;
    for (int c = 0; c < CW; ++c) acc += hp[c] * p[c];
    float s = acc / nrm[0];
    score[i] = s;
    key[i]   = f2ord(s);
}

__global__ void pool_init(unsigned* prefix, int* kneed, int* tiec, int* outc, int k) {
    int g = threadIdx.x;
    if (g < NB) { prefix[g] = 0u; kneed[g] = k; tiec[g] = 0; outc[g] = 0; }
}

__global__ void hist_pass(const unsigned* __restrict__ key, int n, int npg, int shift,
                          unsigned hmask, const unsigned* __restrict__ prefix, int* hist) {
    int i = blockIdx.x * blockDim.x + threadIdx.x;
    if (i >= n) return;
    int g = i / npg;
    unsigned k = key[i];
    if (hmask && ((k ^ prefix[g]) & hmask)) return;
    atomicAdd(&hist[g * 256 + ((k >> shift) & 255u)], 1);
}

__global__ void scan_pass(int* hist, unsigned* prefix, int* kneed, int shift) {
    int g = threadIdx.x;
    if (g >= NB) return;
    int rem = kneed[g];
    int chosen = 0;
    for (int d = 255; d >= 0; --d) {
        int c = hist[g * 256 + d];
        if (c >= rem) { chosen = d; break; }
        rem -= c;
    }
    prefix[g] |= ((unsigned)chosen) << shift;
    kneed[g] = rem;
    for (int d = 0; d < 256; ++d) hist[g * 256 + d] = 0;
}

__global__ void select_nodes(const unsigned* __restrict__ key, const float* __restrict__ score,
                             int n, int npg, int k, const unsigned* __restrict__ prefix,
                             const int* __restrict__ kneed, int* tiec, int* outc,
                             int* o2n, int* perm, float* scale) {
    int i = blockIdx.x * blockDim.x + threadIdx.x;
    if (i >= n) return;
    int g = i / npg;
    unsigned T = prefix[g], kk = key[i];
    bool keep = false;
    if (kk > T) keep = true;
    else if (kk == T) { if (atomicAdd(&tiec[g], 1) < kneed[g]) keep = true; }
    if (keep) {
        int pos = atomicAdd(&outc[g], 1);
        int nn = g * k + pos;
        o2n[i] = nn;
        perm[nn] = i;
        scale[nn] = tanhf(score[i]);
    }
}

__global__ void gather_scale(const float* __restrict__ h, const int* __restrict__ perm,
                             const float* __restrict__ scale, float* o, int m) {
    int t = blockIdx.x * blockDim.x + threadIdx.x;
    if (t >= m * CW) return;
    int nn = t >> 6, c = t & 63;
    o[t] = h[(size_t)perm[nn] * CW + c] * scale[nn];
}

__global__ void edge_compact(const int* __restrict__ src, const int* __restrict__ dst, int ne,
                             const int* __restrict__ o2n, int* nsrc, int* ndst, int* ecnt) {
    int e = blockIdx.x * blockDim.x + threadIdx.x;
    if (e >= ne) return;
    int a = o2n[src[e]], b = o2n[dst[e]];
    if (a >= 0 && b >= 0) {
        int p = atomicAdd(ecnt, 1);
        nsrc[p] = a;
        ndst[p] = b;
    }
}

__global__ void readout(const float* __restrict__ h, int npg, float* xo) {
    int g = blockIdx.x, c = threadIdx.x;   // 8 blocks x 64 threads
    const float* base = h + (size_t)g * npg * CW + c;
    float mx = -3.402823466e38f, sm = 0.f;
    for (int j = 0; j < npg; ++j) {
        float v = base[(size_t)j * CW];
        mx = fmaxf(mx, v);
        sm += v;
    }
    xo[g * 128 + c]      = mx;
    xo[g * 128 + 64 + c] = sm / (float)npg;
}

// ---------------- SAGE2 aggregation (64-ch scatter) ----------------
__global__ void sage_scatter(const int* __restrict__ src, const int* __restrict__ dst,
                             const int* __restrict__ necnt, const float* __restrict__ h,
                             float* s, float* cnt, int maxe) {
    int t = blockIdx.x * blockDim.x + threadIdx.x;
    int e = t >> 4;
    if (e >= maxe) return;
    if (e >= *necnt) return;
    int c4 = (t & 15) << 2;
    int a = src[e], d = dst[e];
    const float* hp = h + (size_t)a * CW + c4;
    float* sp = s + (size_t)d * CW + c4;
    atomicAdd(sp + 0, hp[0]);
    atomicAdd(sp + 1, hp[1]);
    atomicAdd(sp + 2, hp[2]);
    atomicAdd(sp + 3, hp[3]);
    if (c4 == 0) atomicAdd(&cnt[d], 1.f);
}

__global__ void mean_div(float* s, const float* __restrict__ cnt, int m) {
    int t = blockIdx.x * blockDim.x + threadIdx.x;
    if (t >= m * CW) return;
    s[t] /= fmaxf(cnt[t >> 6], 1.f);
}

// ---------------- WMMA f32 GEMM: out = act(A@W [+ A2@W2] + b) ----------------
// One wave per 16x16 output tile; K looped in steps of 4 via V_WMMA_F32_16X16X4_F32.
__global__ void wmma_gemm(const float* __restrict__ A, const float* __restrict__ W,
                          const float* __restrict__ A2, const float* __restrict__ W2,
                          const float* __restrict__ bias, float* __restrict__ out,
                          int M, int K, int N, int relu) {
    int wid  = (blockIdx.x * blockDim.x + threadIdx.x) >> 5;
    int lane = threadIdx.x & 31;
    int ntn = N >> 4;
    int row_t = wid / ntn;
    int col_t = wid - row_t * ntn;
    if (row_t * 16 >= M) return;               // wave-uniform: EXEC all-ones at WMMA
    int half = lane >> 4;                      // 0 or 1
    int l15  = lane & 15;
    int m = row_t * 16 + l15;
    int n = col_t * 16 + l15;
    int kh = half * 2;                         // A lanes 0-15: K={0,1}; lanes 16-31: K={2,3}
    v8f acc = {0.f, 0.f, 0.f, 0.f, 0.f, 0.f, 0.f, 0.f};
    for (int k = 0; k < K; k += 4) {
        v2f a, b;
        a.x = A[(size_t)m * K + k + kh];
        a.y = A[(size_t)m * K + k + kh + 1];
        b.x = W[(size_t)(k + kh) * N + n];
        b.y = W[(size_t)(k + kh + 1) * N + n];
        acc = __builtin_amdgcn_wmma_f32_16x16x4_f32(false, a, false, b, (short)0, acc, false, false);
    }
    if (A2) {
        for (int k = 0; k < K; k += 4) {
            v2f a, b;
            a.x = A2[(size_t)m * K + k + kh];
            a.y = A2[(size_t)m * K + k + kh + 1];
            b.x = W2[(size_t)(k + kh) * N + n];
            b.y = W2[(size_t)(k + kh + 1) * N + n];
            acc = __builtin_amdgcn_wmma_f32_16x16x4_f32(false, a, false, b, (short)0, acc, false, false);
        }
    }
    float bv = bias ? bias[n] : 0.f;
#pragma unroll
    for (int r = 0; r < 8; ++r) {              // D: VGPR r -> row r (lanes 0-15) / r+8 (lanes 16-31)
        float v = acc[r] + bv;
        if (relu) v = fmaxf(v, 0.f);
        out[(size_t)(row_t * 16 + half * 8 + r) * N + n] = v;
    }
}

// ---------------- MLP head helpers ----------------
__global__ void build_z(const float* __restrict__ x1, const float* __restrict__ x2, float* z) {
    int t = blockIdx.x * blockDim.x + threadIdx.x;
    if (t >= 16 * 128) return;
    z[t] = ((t >> 7) < 8) ? (x1[t] + x2[t]) : 0.f;   // pad rows 8..15 with zeros for WMMA
}

__global__ void final_head(const float* __restrict__ z3, const float* __restrict__ w,
                           const float* __restrict__ b, float* o) {
    int g = threadIdx.x;
    if (g >= NB) return;
    float acc = b[0];
    for (int c = 0; c < 64; ++c) acc += z3[g * 64 + c] * w[c];
    o[g] = acc;
}

#define LT(n) dim3((unsigned)((((size_t)(n)) + 255) / 256)), dim3(256)

extern "C" void kernel_launch(void* const* d_in, const int* in_sizes, int n_in,
                              void* d_out, int out_size, void* d_ws, size_t ws_size,
                              hipStream_t stream) {
    (void)in_sizes; (void)n_in; (void)out_size; (void)ws_size;
    const float* x   = (const float*)d_in[0];
    const int*   src = (const int*)d_in[1];
    const int*   dst = src + ETOT;
    const float* w1l = (const float*)d_in[2];
    const float* w1r = (const float*)d_in[3];
    const float* b1  = (const float*)d_in[4];
    const float* w2l = (const float*)d_in[5];
    const float* w2r = (const float*)d_in[6];
    const float* b2  = (const float*)d_in[7];
    const float* p1  = (const float*)d_in[13];
    const float* p2  = (const float*)d_in[14];
    const float* lw1 = (const float*)d_in[17];
    const float* lb1 = (const float*)d_in[18];
    const float* lw2 = (const float*)d_in[19];
    const float* lb2 = (const float*)d_in[20];
    const float* lw3 = (const float*)d_in[21];
    const float* lb3 = (const float*)d_in[22];
    float* outp = (float*)d_out;

    // ---- carve workspace (4B elems, 256B aligned) ----
    char* wsb = (char*)d_ws;
    size_t off = 0;
    auto carve = [&](size_t elems) -> char* {
        char* p = wsb + off;
        off += ((elems * 4 + 255) & ~(size_t)255);
        return p;
    };
    float*    s1    = (float*)carve((size_t)NTOT * 2);
    float*    cnt1  = (float*)carve(NTOT);
    float*    h1    = (float*)carve((size_t)NTOT * CW);
    float*    score = (float*)carve(NTOT);
    unsigned* key   = (unsigned*)carve(NTOT);
    float*    nrm   = (float*)carve(64);
    int*      hist  = (int*)carve(NB * 256);
    unsigned* prefix= (unsigned*)carve(64);
    int*      kneed = (int*)carve(64);
    int*      tiec  = (int*)carve(64);
    int*      outc  = (int*)carve(64);
    int*      ecnt  = (int*)carve(64);
    int*      o2n   = (int*)carve(NTOT);
    int*      perm  = (int*)carve(NTOT / 2);
    float*    scale = (float*)carve(NTOT / 2);
    float*    h1p   = (float*)carve((size_t)(NTOT / 2) * CW);
    int*      src2  = (int*)carve(ETOT);
    int*      dst2  = (int*)carve(ETOT);
    float*    s2    = (float*)carve((size_t)(NTOT / 2) * CW);
    float*    cnt2  = (float*)carve(NTOT / 2);
    float*    h2    = (float*)carve((size_t)(NTOT / 2) * CW);
    float*    h2p   = (float*)carve((size_t)(NTOT / 4) * CW);
    float*    x1    = (float*)carve(NB * 128);
    float*    x2    = (float*)carve(NB * 128);
    float*    z     = (float*)carve(16 * 128);
    float*    z2    = (float*)carve(16 * 128);
    float*    z3    = (float*)carve(16 * 64);

    auto launch_gemm = [&](const float* A, const float* W, const float* A2, const float* W2,
                           const float* bias, float* o, int M, int K, int N, int relu) {
        int waves = (M / 16) * (N / 16);
        int thr = waves * 32;
        int blocks = (thr + 255) / 256;
        wmma_gemm<<<dim3(blocks), dim3(256), 0, stream>>>(A, W, A2, W2, bias, o, M, K, N, relu);
    };

    auto run_pool = [&](const float* h, const float* pvec, int n, int npg,
                        int* permO, float* scaleO, float* hp, float* xo) {
        int k = npg / 2;
        vec_norm64<<<1, 64, 0, stream>>>(pvec, nrm);
        score_kernel<<<LT(n), 0, stream>>>(h, pvec, nrm, score, key, n);
        fill_i32<<<LT(NB * 256), 0, stream>>>(hist, 0, NB * 256);
        pool_init<<<1, 32, 0, stream>>>(prefix, kneed, tiec, outc, k);
        for (int p = 0; p < 4; ++p) {
            int shift = 24 - 8 * p;
            unsigned hmask = (p == 0) ? 0u : (0xFFFFFFFFu << (shift + 8));
            hist_pass<<<LT(n), 0, stream>>>(key, n, npg, shift, hmask, prefix, hist);
            scan_pass<<<1, 32, 0, stream>>>(hist, prefix, kneed, shift);
        }
        fill_i32<<<LT(n), 0, stream>>>(o2n, -1, n);
        select_nodes<<<LT(n), 0, stream>>>(key, score, n, npg, k, prefix, kneed, tiec, outc,
                                           o2n, permO, scaleO);
        int m = n / 2;
        gather_scale<<<LT((size_t)m * CW), 0, stream>>>(h, permO, scaleO, hp, m);
        readout<<<dim3(NB), dim3(64), 0, stream>>>(hp, k, xo);
    };

    // ===== SAGE1: h1 = relu(mean_agg @ w1l + x @ w1r + b1) =====
    fill_f32<<<LT((size_t)NTOT * 2), 0, stream>>>(s1, 0.f, NTOT * 2);
    fill_f32<<<LT(NTOT), 0, stream>>>(cnt1, 0.f, NTOT);
    edge_scatter_dim2<<<LT(ETOT), 0, stream>>>(src, dst, x, s1, cnt1, ETOT);
    sage1_kernel<<<LT((size_t)NTOT * CW), 0, stream>>>(x, s1, cnt1, w1l, w1r, b1, h1, NTOT);

    // ===== pool1 (+ readout x1) and edge compaction =====
    run_pool(h1, p1, NTOT, NPER, perm, scale, h1p, x1);
    fill_i32<<<LT(1), 0, stream>>>(ecnt, 0, 1);
    edge_compact<<<LT(ETOT), 0, stream>>>(src, dst, ETOT, o2n, src2, dst2, ecnt);

    // ===== SAGE2: h2 = relu(mean @ w2l + h1p @ w2r + b2)  [WMMA GEMM] =====
    const int M2 = NTOT / 2;  // 32768
    fill_f32<<<LT((size_t)M2 * CW), 0, stream>>>(s2, 0.f, M2 * CW);
    fill_f32<<<LT(M2), 0, stream>>>(cnt2, 0.f, M2);
    sage_scatter<<<LT((size_t)ETOT * 16), 0, stream>>>(src2, dst2, ecnt, h1p, s2, cnt2, ETOT);
    mean_div<<<LT((size_t)M2 * CW), 0, stream>>>(s2, cnt2, M2);
    launch_gemm(s2, w2l, h1p, w2r, b2, h2, M2, CW, CW, 1);

    // ===== pool2 (+ readout x2) — edges after this are dead code =====
    run_pool(h2, p2, M2, NPER / 2, perm, scale, h2p, x2);

    // ===== MLP head on z = x1 + x2 (rows padded to 16 for WMMA) =====
    build_z<<<LT(16 * 128), 0, stream>>>(x1, x2, z);
    launch_gemm(z,  lw1, nullptr, nullptr, lb1, z2, 16, 128, 128, 1);
    launch_gemm(z2, lw2, nullptr, nullptr, lb2, z3, 16, 128, 64, 1);
    final_head<<<1, 32, 0, stream>>>(z3, lw3, lb3, outp);
}